// CrossActivationRegularization_52879637348367
// MI455X (gfx1250) — compile-verified
//
#include <hip/hip_runtime.h>
#include <cstdint>
#include <cstddef>

// Problem constants (from reference)
#define F_DIM   128
#define T_DIM   4096
#define B_DIM   64
#define TSPLIT  8                       // T-chunks per batch -> 64*8 = 512 blocks
#define CHUNK_T (T_DIM / TSPLIT)        // 512 timesteps per block
#define KT      32                      // timesteps staged in LDS per double-buffer slot
#define NCH     (CHUNK_T / KT)          // 16 chunks per block
#define LDW     (F_DIM + 16)            // padded LDS row: half-waves hit disjoint bank sets
#define NPAIRS  (F_DIM * (F_DIM - 1) / 2)   // 8128

typedef __attribute__((ext_vector_type(2))) float v2f;
typedef __attribute__((ext_vector_type(8))) float v8f;

__device__ __forceinline__ void atomic_add_f32(float* p, float v) {
  __hip_atomic_fetch_add(p, v, __ATOMIC_RELAXED, __HIP_MEMORY_SCOPE_AGENT);
}

// CDNA5 async global->LDS copy (GVS mode: SGPR64 base + per-lane 32-bit offset).
// Tracked by ASYNCcnt; LDS destination address per lane from VGPR.
__device__ __forceinline__ void async_b128(unsigned lds_addr, unsigned goff, const float* base) {
  asm volatile("global_load_async_to_lds_b128 %0, %1, %2"
               :: "v"(lds_addr), "v"(goff), "s"(base)
               : "memory");
}

// ---------------------------------------------------------------------------
// Kernel A: per (batch, T-chunk) partial Gram (raw E[x x^T] * T), column sums
// and sum(|x|), accumulated to workspace with f32 atomics.
// Block = 256 threads = 8 waves; wave w owns output row-tile w (16 rows) and
// computes all 8 column tiles with V_WMMA_F32_16X16X4_F32.
// ---------------------------------------------------------------------------
__global__ __launch_bounds__(256, 1)
void syrk_partial_kernel(const float* __restrict__ x,
                         float* __restrict__ gram,     // [B][F][F]
                         float* __restrict__ colsum,   // [B][F]
                         float* __restrict__ abssum)   // [1]
{
  __shared__ float xs[2][KT][LDW];     // double-buffered tile of x: [time][filter]
  __shared__ float cred[8][32][4];     // column-sum cross-wave reduction
  __shared__ float ared[256];          // |x| reduction

  const int tid  = threadIdx.x;
  const int wave = tid >> 5;           // 0..7  == output row-tile, also loader row-group
  const int lane = tid & 31;
  const int m    = lane & 15;          // matrix row/col within tile
  const int kh   = lane >> 4;          // half-wave selects K pair (ISA 16x4 A layout)
  const int b    = blockIdx.x;
  const int tc   = blockIdx.y;

  const float* xb = x + ((size_t)b * T_DIM + (size_t)tc * CHUNK_T) * (size_t)F_DIM;

  v8f acc[8] = {};                     // 8 tiles x 16x16 f32 accumulators

  float s0 = 0.f, s1 = 0.f, s2 = 0.f, s3 = 0.f, aab = 0.f;

  // Prologue: async-stage chunk 0 into buffer 0.
  #pragma unroll
  for (int p = 0; p < 4; ++p) {
    const int row = wave + 8 * p;
    const unsigned lds  = (unsigned)(size_t)&xs[0][row][4 * lane];
    const unsigned goff = (unsigned)(((row) * F_DIM + 4 * lane) * sizeof(float));
    async_b128(lds, goff, xb);
  }

  for (int ci = 0; ci < NCH; ++ci) {
    const int buf = ci & 1;

    if (ci + 1 < NCH) {
      // Prefetch next chunk into the other buffer while this one is consumed.
      #pragma unroll
      for (int p = 0; p < 4; ++p) {
        const int row = wave + 8 * p;
        const unsigned lds  = (unsigned)(size_t)&xs[buf ^ 1][row][4 * lane];
        const unsigned goff = (unsigned)((((ci + 1) * KT + row) * F_DIM + 4 * lane) * sizeof(float));
        async_b128(lds, goff, xb);
      }
      // 4 newer async ops may stay in flight; the 4 for chunk ci must be done.
      asm volatile("s_wait_asynccnt 4" ::: "memory");
    } else {
      asm volatile("s_wait_asynccnt 0" ::: "memory");
    }
    __syncthreads();   // publish this wave's LDS writes to the whole block

    // ---- column sums + L1 piggyback (each thread re-reads the 4 filters it staged)
    #pragma unroll
    for (int p = 0; p < 4; ++p) {
      const int row = wave + 8 * p;
      const float* r = &xs[buf][row][4 * lane];
      const float v0 = r[0], v1 = r[1], v2 = r[2], v3 = r[3];
      s0 += v0; s1 += v1; s2 += v2; s3 += v3;
      aab += fabsf(v0) + fabsf(v1) + fabsf(v2) + fabsf(v3);
    }

    // ---- WMMA: Gram += X_tile^T X_tile, K stepped by 4
    #pragma unroll
    for (int k0 = 0; k0 < KT; k0 += 4) {
      v2f a;   // A 16x4: lane m, half-wave kh -> K = 2*kh + {0,1} (ISA table)
      a.x = xs[buf][k0 + 2 * kh + 0][wave * 16 + m];
      a.y = xs[buf][k0 + 2 * kh + 1][wave * 16 + m];
      #pragma unroll
      for (int nt = 0; nt < 8; ++nt) {
        v2f bm;  // B 4x16 assumed mirrored layout: row K = 2*kh + v, col = m
        bm.x = xs[buf][k0 + 2 * kh + 0][nt * 16 + m];
        bm.y = xs[buf][k0 + 2 * kh + 1][nt * 16 + m];
        acc[nt] = __builtin_amdgcn_wmma_f32_16x16x4_f32(
            false, a, false, bm, (short)0, acc[nt], false, false);
      }
    }
    __syncthreads();   // everyone done reading buf before it is refilled
  }

  // ---- reduce column sums across the 8 row-groups, then atomically merge
  cred[wave][lane][0] = s0; cred[wave][lane][1] = s1;
  cred[wave][lane][2] = s2; cred[wave][lane][3] = s3;
  ared[tid] = aab;
  __syncthreads();
  if (wave == 0) {
    float t0 = 0.f, t1 = 0.f, t2 = 0.f, t3 = 0.f;
    #pragma unroll
    for (int gg = 0; gg < 8; ++gg) {
      t0 += cred[gg][lane][0]; t1 += cred[gg][lane][1];
      t2 += cred[gg][lane][2]; t3 += cred[gg][lane][3];
    }
    atomic_add_f32(&colsum[b * F_DIM + 4 * lane + 0], t0);
    atomic_add_f32(&colsum[b * F_DIM + 4 * lane + 1], t1);
    atomic_add_f32(&colsum[b * F_DIM + 4 * lane + 2], t2);
    atomic_add_f32(&colsum[b * F_DIM + 4 * lane + 3], t3);
  }
  for (int off = 128; off > 0; off >>= 1) {
    if (tid < off) ared[tid] += ared[tid + off];
    __syncthreads();
  }
  if (tid == 0) atomic_add_f32(abssum, ared[0]);

  // ---- merge partial Gram tiles (C layout: VGPR r, lane -> M = r + 8*kh, N = m)
  #pragma unroll
  for (int nt = 0; nt < 8; ++nt) {
    #pragma unroll
    for (int r = 0; r < 8; ++r) {
      const int row = wave * 16 + r + 8 * kh;
      const int col = nt * 16 + m;
      atomic_add_f32(&gram[((size_t)b * F_DIM + row) * F_DIM + col], acc[nt][r]);
    }
  }
}

// ---------------------------------------------------------------------------
// Kernel B1: per (b, f) mean and 1/std from Gram diagonal + column sums.
// ---------------------------------------------------------------------------
__global__ __launch_bounds__(256)
void stats_kernel(const float* __restrict__ gram, const float* __restrict__ colsum,
                  float* __restrict__ meanv, float* __restrict__ invstd)
{
  const int i = blockIdx.x * 256 + threadIdx.x;     // 0 .. B*F-1
  const int b = i >> 7, f = i & (F_DIM - 1);
  const float mean = colsum[i] * (1.0f / T_DIM);
  const float var  = gram[((size_t)b * F_DIM + f) * F_DIM + f] * (1.0f / T_DIM) - mean * mean;
  meanv[i]  = mean;
  invstd[i] = rsqrtf(var);
}

// ---------------------------------------------------------------------------
// Kernel B2: tri_sum = sum_{g>f} | mean_b corr[b,f,g] |
// ---------------------------------------------------------------------------
__global__ __launch_bounds__(256)
void corr_reduce_kernel(const float* __restrict__ gram, const float* __restrict__ meanv,
                        const float* __restrict__ invstd, float* __restrict__ tri)
{
  __shared__ float red[256];
  const int i = blockIdx.x * 256 + threadIdx.x;     // 0 .. F*F-1
  const int f = i >> 7, g = i & (F_DIM - 1);
  float s = 0.f;
  for (int b = 0; b < B_DIM; ++b) {
    const float cov = gram[((size_t)b * F_DIM + f) * F_DIM + g] * (1.0f / T_DIM)
                    - meanv[b * F_DIM + f] * meanv[b * F_DIM + g];
    s += cov * invstd[b * F_DIM + f] * invstd[b * F_DIM + g];
  }
  red[threadIdx.x] = (g > f) ? fabsf(s * (1.0f / B_DIM)) : 0.f;
  __syncthreads();
  for (int off = 128; off > 0; off >>= 1) {
    if (threadIdx.x < off) red[threadIdx.x] += red[threadIdx.x + off];
    __syncthreads();
  }
  if (threadIdx.x == 0) atomic_add_f32(tri, red[0]);
}

// ---------------------------------------------------------------------------
// Kernel C: final 3-element output.
// ---------------------------------------------------------------------------
__global__ void finalize_kernel(const float* __restrict__ tri,
                                const float* __restrict__ abssum,
                                float* __restrict__ out)
{
  if (threadIdx.x == 0) {
    const float t = *tri;
    out[0] = t * (0.01f / (float)NPAIRS);   // regularization loss
    out[1] = t;                             // cross-activation tri-sum
    out[2] = *abssum * (1.0f / F_DIM);      // activity L1
  }
}

extern "C" void kernel_launch(void* const* d_in, const int* in_sizes, int n_in,
                              void* d_out, int out_size, void* d_ws, size_t ws_size,
                              hipStream_t stream) {
  (void)in_sizes; (void)n_in; (void)out_size; (void)ws_size;
  const float* x = (const float*)d_in[0];
  float* out = (float*)d_out;

  // Workspace layout (floats): gram | colsum | mean | invstd | abssum | tri
  float* ws      = (float*)d_ws;
  float* gram    = ws;                                       // B*F*F = 1048576
  float* colsum  = gram   + (size_t)B_DIM * F_DIM * F_DIM;   // B*F = 8192
  float* meanv   = colsum + (size_t)B_DIM * F_DIM;           // 8192
  float* invstd  = meanv  + (size_t)B_DIM * F_DIM;           // 8192
  float* abssum  = invstd + (size_t)B_DIM * F_DIM;           // 1
  float* tri     = abssum + 1;                               // 1

  // Zero accumulation buffers every call (workspace is not re-poisoned).
  hipMemsetAsync(gram, 0, ((size_t)B_DIM * F_DIM * F_DIM + (size_t)B_DIM * F_DIM) * sizeof(float), stream);
  hipMemsetAsync(abssum, 0, 2 * sizeof(float), stream);

  dim3 gridA(B_DIM, TSPLIT);
  syrk_partial_kernel<<<gridA, 256, 0, stream>>>(x, gram, colsum, abssum);
  stats_kernel<<<(B_DIM * F_DIM) / 256, 256, 0, stream>>>(gram, colsum, meanv, invstd);
  corr_reduce_kernel<<<(F_DIM * F_DIM) / 256, 256, 0, stream>>>(gram, meanv, invstd, tri);
  finalize_kernel<<<1, 32, 0, stream>>>(tri, abssum, out);
}